// GATEncoder_11012296147174
// MI455X (gfx1250) — compile-verified
//
#include <hip/hip_runtime.h>

// ---------------------------------------------------------------------------
// GAT encoder for MI455X (gfx1250, wave32).
//  - Dense GEMMs: v_wmma_f32_16x16x4_f32, one wave per 16-row strip with
//    register blocking across all N tiles (A read exactly once), B staged
//    through LDS via gfx1250 async-to-LDS copies (global_load_async_to_lds_b128
//    + s_wait_asynccnt), fed to WMMA as ds_load fragments.
//  - Edge softmax: flip-encoded u32 atomicMax + f32 atomicAdd (L2-resident).
//  - Aggregation: one wave per (edge, head), lanes = channels (coalesced
//    gathers and coalesced global_atomic_add_f32).
// ---------------------------------------------------------------------------

typedef __attribute__((ext_vector_type(2))) float v2f;
typedef __attribute__((ext_vector_type(8))) float v8f;

#define ACT_NONE  0
#define ACT_LEAKY 1

__device__ __forceinline__ float leakyf(float v, float slope) {
  return v >= 0.f ? v : slope * v;
}

// Order-preserving float -> uint mapping for atomicMax-based segment max.
__device__ __forceinline__ unsigned flipf(float x) {
  unsigned u = __float_as_uint(x);
  return (u & 0x80000000u) ? ~u : (u | 0x80000000u);
}
__device__ __forceinline__ float unflipf(unsigned u) {
  return __uint_as_float((u & 0x80000000u) ? (u & 0x7FFFFFFFu) : ~u);
}

// Async copy 16 bytes global -> LDS (gfx1250 VGLOBAL async op, ASYNCcnt).
__device__ __forceinline__ void async_b128_to_lds(unsigned lds_off,
                                                  const float* gptr) {
  asm volatile("global_load_async_to_lds_b128 %0, %1, off"
               :: "v"(lds_off), "v"(gptr) : "memory");
}
__device__ __forceinline__ void wait_async_all() {
  asm volatile("s_wait_asynccnt 0x0" ::: "memory");
}

// ---------------------------------------------------------------------------
// C[:, col0 : col0+NT*16] = act(A[MxK] @ B[K x NT*16] + bias)
// One wave per 16-row strip; NT accumulators (full N range per wave) so each
// A element is loaded exactly once. B K-chunks staged in LDS via async DMA.
// Requirements: M % 16 == 0, K % 16 == 0.
// ---------------------------------------------------------------------------
constexpr int KC = 16;  // K-chunk staged per round

template <int NT>
__global__ void wmma_gemm_nt(const float* __restrict__ A,
                             const float* __restrict__ B,
                             const float* __restrict__ bias,
                             float* __restrict__ C,
                             int M, int K, int ldC, int col0,
                             int act, float slope) {
  constexpr int Nc = NT * 16;
  __shared__ float sB[KC * Nc];

  const int lane    = threadIdx.x & 31;
  const int wave    = threadIdx.x >> 5;
  const int waves   = blockDim.x >> 5;
  const int tiles_m = M >> 4;
  const int rowtile = blockIdx.x * waves + wave;
  const bool active = rowtile < tiles_m;   // wave-uniform

  // A fragment: 16x4 f32. lane L: row = L&15; lanes<16 hold K+0/K+1,
  // lanes>=16 hold K+2/K+3 (VGPR0/VGPR1).
  const int n15   = lane & 15;
  const int khalf = (lane >> 4) << 1;      // 0 or 2
  const int mrow  = (rowtile << 4) + n15;

  const float* __restrict__ arow = A + (long long)(active ? mrow : 0) * K;

  v8f acc[NT];
#pragma unroll
  for (int t = 0; t < NT; ++t) acc[t] = (v8f){};

  for (int kc = 0; kc < K; kc += KC) {
    __syncthreads();   // previous chunk fully consumed by all waves
    // Stage B[kc:kc+KC, :] -- contiguous KC*Nc floats -- into LDS via async DMA.
    const float* bchunk = B + (long long)kc * Nc;
    for (int i = threadIdx.x; i < (KC * Nc) / 4; i += blockDim.x) {
      unsigned lds_off = (unsigned)(uintptr_t)(&sB[i * 4]);
      async_b128_to_lds(lds_off, bchunk + i * 4);
    }
    wait_async_all();
    __syncthreads();

    if (active) {   // wave-uniform branch: EXEC all-ones inside for WMMA
#pragma unroll
      for (int k4 = 0; k4 < KC; k4 += 4) {
        __builtin_prefetch(arow + kc + k4 + 64, 0, 0);  // global_prefetch_b8
        v2f a;
        a.x = arow[kc + k4 + khalf];
        a.y = arow[kc + k4 + khalf + 1];
#pragma unroll
        for (int t = 0; t < NT; ++t) {
          v2f b;   // B fragment mirrors A: lane halves split K, cols on lanes
          b.x = sB[(k4 + khalf)     * Nc + t * 16 + n15];
          b.y = sB[(k4 + khalf + 1) * Nc + t * 16 + n15];
          acc[t] = __builtin_amdgcn_wmma_f32_16x16x4_f32(
              false, a, false, b, (short)0, acc[t], false, false);
        }
      }
    }
  }

  if (active) {
    // C/D layout: lane L, vgpr r -> row = 8*(L>>4)+r, col = L&15 (in tile).
    const int rbase = (rowtile << 4) + ((lane >> 4) << 3);
#pragma unroll
    for (int t = 0; t < NT; ++t) {
      const int  ccol = col0 + t * 16 + n15;
      const float bv  = bias ? bias[t * 16 + n15] : 0.f;
#pragma unroll
      for (int r = 0; r < 8; ++r) {
        float v = acc[t][r] + bv;
        if (act == ACT_LEAKY) v = leakyf(v, slope);
        C[(long long)(rbase + r) * ldC + ccol] = v;
      }
    }
  }
}

// Tiny-K dense encoder (K = 5 or 1): one thread per output element.
__global__ void small_gemm_leaky(const float* __restrict__ A,
                                 const float* __restrict__ B,
                                 const float* __restrict__ bias,
                                 float* __restrict__ C,
                                 int M, int K, int Nc, int ldC, int col0,
                                 float slope) {
  int tid = blockIdx.x * blockDim.x + threadIdx.x;
  if (tid >= M * Nc) return;
  int n = tid / Nc, c = tid - n * Nc;
  float acc = bias[c];
  for (int k = 0; k < K; ++k) acc += A[(long long)n * K + k] * B[k * Nc + c];
  C[(long long)n * ldC + col0 + c] = leakyf(acc, slope);
}

// Per-node attention logits: alpha_s/alpha_d = <h[n,h,:], a_src/a_dst[h,:]>.
__global__ void node_alpha(const float* __restrict__ h,
                           const float* __restrict__ a_src,
                           const float* __restrict__ a_dst,
                           float* __restrict__ as, float* __restrict__ ad,
                           int n_nodes, int H, int C) {
  int tid = blockIdx.x * blockDim.x + threadIdx.x;
  if (tid >= n_nodes * H) return;
  int n = tid / H, hh = tid - n * H;
  const float* hv  = h + (long long)n * H * C + hh * C;
  const float* asw = a_src + hh * C;
  const float* adw = a_dst + hh * C;
  float s = 0.f, d = 0.f;
  for (int c = 0; c < C; ++c) { float x = hv[c]; s += x * asw[c]; d += x * adw[c]; }
  as[tid] = s;
  ad[tid] = d;
}

// Init accumulator with bias, reset segment max/denominator.
__global__ void init_layer(float* __restrict__ out, const float* __restrict__ bias,
                           unsigned* __restrict__ mbuf, float* __restrict__ denom,
                           int n_nodes, int H, int C) {
  int n = blockIdx.x * blockDim.x + threadIdx.x;
  if (n >= n_nodes) return;
  for (int hh = 0; hh < H; ++hh) { mbuf[n * H + hh] = 0u; denom[n * H + hh] = 0.f; }
  int HC = H * C;
  float* o = out + (long long)n * HC;
  for (int c = 0; c < HC; ++c) o[c] = bias[c];
}

__device__ __forceinline__ void edge_nodes(const int* src, const int* dst,
                                           int E, int e, int& s, int& d) {
  if (e < E) { s = src[e]; d = dst[e]; }
  else       { s = d = e - E; }        // appended self-loops
}

__global__ void edge_max(const int* __restrict__ src, const int* __restrict__ dst,
                         int E, int n_nodes,
                         const float* __restrict__ as, const float* __restrict__ ad,
                         unsigned* __restrict__ mbuf, int H) {
  int tid = blockIdx.x * blockDim.x + threadIdx.x;
  int ET = E + n_nodes;
  if (tid >= ET * H) return;
  int e = tid / H, hh = tid - e * H;
  int s, d; edge_nodes(src, dst, E, e, s, d);
  float v = leakyf(as[s * H + hh] + ad[d * H + hh], 0.2f);
  atomicMax(&mbuf[d * H + hh], flipf(v));
}

__global__ void edge_sum(const int* __restrict__ src, const int* __restrict__ dst,
                         int E, int n_nodes,
                         const float* __restrict__ as, const float* __restrict__ ad,
                         const unsigned* __restrict__ mbuf,
                         float* __restrict__ denom, int H) {
  int tid = blockIdx.x * blockDim.x + threadIdx.x;
  int ET = E + n_nodes;
  if (tid >= ET * H) return;
  int e = tid / H, hh = tid - e * H;
  int s, d; edge_nodes(src, dst, E, e, s, d);
  float v  = leakyf(as[s * H + hh] + ad[d * H + hh], 0.2f);
  float mx = unflipf(mbuf[d * H + hh]);
  atomicAdd(&denom[d * H + hh], __expf(v - mx));
}

// One wave per (edge, head); lanes cover contiguous channels -> coalesced
// gathers of h[src] and coalesced global_atomic_add_f32 to out[dst].
__global__ void edge_scatter(const int* __restrict__ src, const int* __restrict__ dst,
                             int E, int n_nodes,
                             const float* __restrict__ as, const float* __restrict__ ad,
                             const unsigned* __restrict__ mbuf,
                             const float* __restrict__ denom,
                             const float* __restrict__ h,
                             float* __restrict__ out, int H, int C) {
  int gtid = blockIdx.x * blockDim.x + threadIdx.x;
  int wid  = gtid >> 5;                // (edge, head) index
  int lane = gtid & 31;
  int ET = E + n_nodes;
  if (wid >= ET * H) return;
  int e = wid / H, hh = wid - e * H;
  int s, d; edge_nodes(src, dst, E, e, s, d);
  float v     = leakyf(as[s * H + hh] + ad[d * H + hh], 0.2f);
  float mx    = unflipf(mbuf[d * H + hh]);
  float alpha = __expf(v - mx) / (denom[d * H + hh] + 1e-16f);
  const float* hs = h   + (long long)s * H * C + hh * C;
  float*       od = out + (long long)d * H * C + hh * C;
  for (int c = lane; c < C; c += 32) atomicAdd(&od[c], alpha * hs[c]);
}

__global__ void relu_inplace(float* __restrict__ x, int n) {
  int tid = blockIdx.x * blockDim.x + threadIdx.x;
  if (tid < n) x[tid] = fmaxf(x[tid], 0.f);
}

// ---------------------------------------------------------------------------

static inline int cdiv(int a, int b) { return (a + b - 1) / b; }

extern "C" void kernel_launch(void* const* d_in, const int* in_sizes, int n_in,
                              void* d_out, int out_size, void* d_ws, size_t ws_size,
                              hipStream_t stream) {
  const float* des      = (const float*)d_in[0];
  const float* tweet    = (const float*)d_in[1];
  const float* num_prop = (const float*)d_in[2];
  const float* cat_prop = (const float*)d_in[3];
  const int*   eidx     = (const int*)d_in[4];
  const float* W_des    = (const float*)d_in[5];
  const float* b_des    = (const float*)d_in[6];
  const float* W_tweet  = (const float*)d_in[7];
  const float* b_tweet  = (const float*)d_in[8];
  const float* W_num    = (const float*)d_in[9];
  const float* b_num    = (const float*)d_in[10];
  const float* W_cat    = (const float*)d_in[11];
  const float* b_cat    = (const float*)d_in[12];
  const float* W_in     = (const float*)d_in[13];
  const float* b_in     = (const float*)d_in[14];
  const float* W1       = (const float*)d_in[15];
  const float* a_src1   = (const float*)d_in[16];
  const float* a_dst1   = (const float*)d_in[17];
  const float* b1       = (const float*)d_in[18];
  const float* W2       = (const float*)d_in[19];
  const float* a_src2   = (const float*)d_in[20];
  const float* a_dst2   = (const float*)d_in[21];
  const float* b2       = (const float*)d_in[22];

  const int N  = in_sizes[0] / 768;     // 100000
  const int E  = in_sizes[4] / 2;       // 1600000
  const int ET = E + N;                 // with self-loops
  const int HID = 128, OUTC = 64;
  const int* src = eidx;
  const int* dst = eidx + E;

  // Workspace layout (aliased through the pipeline):
  //   bufA: xcat                       [N,128]
  //   bufB: x -> out1 -> x2 (in place) [N,128]
  //   bufC: h1 -> h2                   [N,128]
  float* bufA = (float*)d_ws;
  float* bufB = bufA + (size_t)N * HID;
  float* bufC = bufB + (size_t)N * HID;
  float*    as1  = bufC + (size_t)N * HID;          // [N,4]
  float*    ad1  = as1 + (size_t)N * 4;
  float*    den1 = ad1 + (size_t)N * 4;
  unsigned* m1   = (unsigned*)(den1 + (size_t)N * 4);
  float*    as2  = (float*)(m1 + (size_t)N * 4);    // [N,1]
  float*    ad2  = as2 + N;
  float*    den2 = ad2 + N;
  unsigned* m2   = (unsigned*)(den2 + N);
  float*    z    = (float*)d_out;                   // [N,64]

  const int TB  = 256;                 // scalar-kernel block
  const int GB  = 256;                 // 8 waves per WMMA block
  const int WPB = GB / 32;
  const int gblocks = cdiv(N / 16, WPB);   // one wave per 16-row strip

  // ---- encoders -> xcat (bufA) ----
  wmma_gemm_nt<2><<<gblocks, GB, 0, stream>>>(des,   W_des,   b_des,   bufA,
      N, 768, HID, 0,  ACT_LEAKY, 0.01f);
  wmma_gemm_nt<2><<<gblocks, GB, 0, stream>>>(tweet, W_tweet, b_tweet, bufA,
      N, 768, HID, 32, ACT_LEAKY, 0.01f);
  small_gemm_leaky<<<cdiv(N * 32, TB), TB, 0, stream>>>(num_prop, W_num, b_num, bufA,
      N, 5, 32, HID, 64, 0.01f);
  small_gemm_leaky<<<cdiv(N * 32, TB), TB, 0, stream>>>(cat_prop, W_cat, b_cat, bufA,
      N, 1, 32, HID, 96, 0.01f);

  // ---- x = leaky(xcat @ W_in + b_in) -> bufB ----
  wmma_gemm_nt<8><<<gblocks, GB, 0, stream>>>(bufA, W_in, b_in, bufB,
      N, HID, HID, 0, ACT_LEAKY, 0.01f);

  // ---- GAT layer 1: h1 = x @ W1 -> bufC ----
  wmma_gemm_nt<8><<<gblocks, GB, 0, stream>>>(bufB, W1, nullptr, bufC,
      N, HID, HID, 0, ACT_NONE, 0.f);
  node_alpha<<<cdiv(N * 4, TB), TB, 0, stream>>>(bufC, a_src1, a_dst1, as1, ad1, N, 4, 32);
  init_layer<<<cdiv(N, TB), TB, 0, stream>>>(bufB, b1, m1, den1, N, 4, 32);  // x dead now
  edge_max<<<cdiv(ET * 4, TB), TB, 0, stream>>>(src, dst, E, N, as1, ad1, m1, 4);
  edge_sum<<<cdiv(ET * 4, TB), TB, 0, stream>>>(src, dst, E, N, as1, ad1, m1, den1, 4);
  edge_scatter<<<cdiv(ET * 4 * 32, TB), TB, 0, stream>>>(src, dst, E, N, as1, ad1, m1, den1,
                                                         bufC, bufB, 4, 32);
  relu_inplace<<<cdiv(N * HID, TB), TB, 0, stream>>>(bufB, N * HID);  // x2 = relu(out1)

  // ---- GAT layer 2: h2 = x2 @ W2 -> bufC (h1 dead) ----
  wmma_gemm_nt<4><<<gblocks, GB, 0, stream>>>(bufB, W2, nullptr, bufC,
      N, HID, OUTC, 0, ACT_NONE, 0.f);
  node_alpha<<<cdiv(N, TB), TB, 0, stream>>>(bufC, a_src2, a_dst2, as2, ad2, N, 1, OUTC);
  init_layer<<<cdiv(N, TB), TB, 0, stream>>>(z, b2, m2, den2, N, 1, OUTC);
  edge_max<<<cdiv(ET, TB), TB, 0, stream>>>(src, dst, E, N, as2, ad2, m2, 1);
  edge_sum<<<cdiv(ET, TB), TB, 0, stream>>>(src, dst, E, N, as2, ad2, m2, den2, 1);
  edge_scatter<<<cdiv(ET * 32, TB), TB, 0, stream>>>(src, dst, E, N, as2, ad2, m2, den2,
                                                     bufC, z, 1, OUTC);
}